// GumbelSampling_21904333210076
// MI455X (gfx1250) — compile-verified
//
#include <hip/hip_runtime.h>
#include <stdint.h>

// ---------------------------------------------------------------------------
// Gumbel segmented softmax + straight-through one-hot, MI455X (gfx1250).
// 4 streaming passes over E. Segment reductions via native L2 atomics on
// L2-resident group arrays (u64 fixed-point sums, ordered-uint-key maxes:
// deterministic). The group-index stream (read by every pass) is staged
// into LDS per-workgroup via the Tensor Data Mover (tensor_load_to_lds,
// TENSORcnt), consumed as ds_load_b128.
// ---------------------------------------------------------------------------

#define TEMP_INV   (1.0f / 0.6f)     // 1/TEMPERATURE
#define EPS_NOISE  1e-4f
#define FIX_SCALE  1099511627776.0   // 2^40 fixed-point for segment sums
#define CHUNK      8
#define NTHREADS   256
#define TILE       (NTHREADS * CHUNK)   // 2048 ints = 8KB LDS per block

typedef unsigned int u32x4 __attribute__((ext_vector_type(4)));
typedef int          i32x4 __attribute__((ext_vector_type(4)));
typedef int          i32x8 __attribute__((ext_vector_type(8)));

#if defined(__has_builtin)
#  if __has_builtin(__builtin_amdgcn_tensor_load_to_lds) && \
      __has_builtin(__builtin_amdgcn_s_wait_tensorcnt)
#    define HAVE_TDM 1
#  endif
#endif

// Order-preserving bijection float -> uint32 (a<b <=> key(a)<key(b))
__device__ __forceinline__ unsigned int fkey(float f) {
    unsigned int u = __float_as_uint(f);
    return (u & 0x80000000u) ? ~u : (u | 0x80000000u);
}
__device__ __forceinline__ float funkey(unsigned int k) {
    unsigned int u = (k & 0x80000000u) ? (k & 0x7FFFFFFFu) : ~k;
    return __uint_as_float(u);
}

// ---------------------------------------------------------------------------
// Stage one 2048-int tile of the group-index array into LDS.
// TDM path: 1-D tile descriptor, OOB elements past tensor_dim0 zero-fill
// (hardware tail handling). Issued by wave 0 only; TENSORcnt-tracked.
// ---------------------------------------------------------------------------
__device__ __forceinline__ void stage_grp(const int* __restrict__ grp,
                                          int blockBase, int E,
                                          int* lds_grp) {
#if defined(HAVE_TDM)
    if (threadIdx.x < 32u) {   // wave 0 of the workgroup issues the DMA
        unsigned long long ga = (unsigned long long)(grp + blockBase);
        // flat shared-aperture address: low 32 bits are the LDS byte address
        unsigned lds_off = (unsigned)(unsigned long long)(void*)lds_grp;
        unsigned dim0 = (unsigned)(E - blockBase);   // remaining elements

        // D# group 0: [1:0]=count=1 | [63:32]=lds_addr |
        //             [120:64]=global_addr | [127:126]=type=2
        u32x4 g0 = { 1u,
                     lds_off,
                     (unsigned)ga,
                     ((unsigned)(ga >> 32) & 0x01FFFFFFu) | 0x80000000u };
        // D# group 1: [17:16]=data_size(4B) | [79:48]=tensor_dim0 |
        //             [111:80]=tensor_dim1=1 | [127:112]=tile_dim0=TILE |
        //             tile_dim1=0 (1-D tile) | [207:160]=dim0_stride=TILE
        i32x8 g1 = { (int)(2u << 16),
                     (int)((dim0 & 0xFFFFu) << 16),
                     (int)((dim0 >> 16) | (1u << 16)),
                     (int)((unsigned)TILE << 16),
                     0,
                     (int)TILE,
                     0, 0 };
        i32x4 gz = { 0, 0, 0, 0 };
#if __has_include(<hip/amd_detail/amd_gfx1250_TDM.h>)
        i32x8 gz8 = { 0, 0, 0, 0, 0, 0, 0, 0 };
        __builtin_amdgcn_tensor_load_to_lds(g0, g1, gz, gz, gz8, 0);
#else
        __builtin_amdgcn_tensor_load_to_lds(g0, g1, gz, gz, 0);
#endif
        __builtin_amdgcn_s_wait_tensorcnt(0);
    }
    __syncthreads();
#else
    // Fallback: cooperative coalesced copy
#pragma unroll
    for (int r = 0; r < CHUNK; ++r) {
        int i = blockBase + (int)threadIdx.x + r * NTHREADS;
        lds_grp[threadIdx.x + r * NTHREADS] = (i < E) ? grp[i] : 0;
    }
    __syncthreads();
#endif
}

// ---------------------------------------------------------------------------
// Group-array init / finalize (n_groups only known on device -> grid-stride)
// ---------------------------------------------------------------------------
__global__ void k_init(unsigned int* __restrict__ smax_key,
                       unsigned long long* __restrict__ denom64,
                       unsigned int* __restrict__ m_key,
                       const int* __restrict__ ng_dev) {
    int n = *ng_dev;
    int stride = gridDim.x * blockDim.x;
    for (int i = blockIdx.x * blockDim.x + threadIdx.x; i < n; i += stride) {
        smax_key[i] = 0u;          // below every real key
        denom64[i]  = 0ull;
        m_key[i]    = 0u;
    }
}

__global__ void k_fin_denom(const unsigned long long* __restrict__ denom64,
                            float* __restrict__ denomf,
                            const int* __restrict__ ng_dev) {
    int n = *ng_dev;
    int stride = gridDim.x * blockDim.x;
    for (int i = blockIdx.x * blockDim.x + threadIdx.x; i < n; i += stride) {
        denomf[i] = (float)((double)denom64[i] * (1.0 / FIX_SCALE));
    }
}

// ---------------------------------------------------------------------------
// Pass 1: s = (logits - log(-log(u))) / T ; store s ; segment max (keys)
// ---------------------------------------------------------------------------
__global__ void __launch_bounds__(NTHREADS)
k_pass1(const float* __restrict__ logits,
        const int*   __restrict__ grp,
        const float* __restrict__ ug,
        float* __restrict__ s_out,
        unsigned int* __restrict__ smax_key,
        int E, int use_vec) {
    __shared__ __align__(16) int lds_grp[TILE];
    int blockBase = (int)blockIdx.x * TILE;
    stage_grp(grp, blockBase, E, lds_grp);

    int base = blockBase + (int)threadIdx.x * CHUNK;
    if (base >= E) return;
    const int4* lg = (const int4*)(lds_grp + (int)threadIdx.x * CHUNK);

    if (use_vec && base + CHUNK <= E) {
        float4 la = ((const float4*)(logits + base))[0];
        float4 lb = ((const float4*)(logits + base))[1];
        float4 ua = ((const float4*)(ug + base))[0];
        float4 ub = ((const float4*)(ug + base))[1];
        int4   ga = lg[0];
        int4   gb = lg[1];

        float l[CHUNK] = {la.x, la.y, la.z, la.w, lb.x, lb.y, lb.z, lb.w};
        float u[CHUNK] = {ua.x, ua.y, ua.z, ua.w, ub.x, ub.y, ub.z, ub.w};
        int   g[CHUNK] = {ga.x, ga.y, ga.z, ga.w, gb.x, gb.y, gb.z, gb.w};
        float s[CHUNK];
#pragma unroll
        for (int i = 0; i < CHUNK; ++i)
            s[i] = (l[i] - __logf(-__logf(u[i]))) * TEMP_INV;

        ((float4*)(s_out + base))[0] = make_float4(s[0], s[1], s[2], s[3]);
        ((float4*)(s_out + base))[1] = make_float4(s[4], s[5], s[6], s[7]);

        // run-compressed segment max: sorted groups -> few atomics
        int cg = g[0]; float cm = s[0];
#pragma unroll
        for (int i = 1; i < CHUNK; ++i) {
            if (g[i] == cg) { cm = fmaxf(cm, s[i]); }
            else { atomicMax(&smax_key[cg], fkey(cm)); cg = g[i]; cm = s[i]; }
        }
        atomicMax(&smax_key[cg], fkey(cm));
    } else {
        int end = min(base + CHUNK, E);
        for (int i = base; i < end; ++i) {
            float s = (logits[i] - __logf(-__logf(ug[i]))) * TEMP_INV;
            s_out[i] = s;
            atomicMax(&smax_key[lds_grp[i - blockBase]], fkey(s));
        }
    }
}

// ---------------------------------------------------------------------------
// Pass 2: e = exp(s - smax[g]) ; store e ; segment sum (u64 fixed point)
// ---------------------------------------------------------------------------
__global__ void __launch_bounds__(NTHREADS)
k_pass2(const float* __restrict__ s_in,
        const int*   __restrict__ grp,
        const unsigned int* __restrict__ smax_key,
        float* __restrict__ e_out,
        unsigned long long* __restrict__ denom64,
        int E, int use_vec) {
    __shared__ __align__(16) int lds_grp[TILE];
    int blockBase = (int)blockIdx.x * TILE;
    stage_grp(grp, blockBase, E, lds_grp);

    int base = blockBase + (int)threadIdx.x * CHUNK;
    if (base >= E) return;
    const int4* lg = (const int4*)(lds_grp + (int)threadIdx.x * CHUNK);

    if (use_vec && base + CHUNK <= E) {
        float4 sa = ((const float4*)(s_in + base))[0];
        float4 sb = ((const float4*)(s_in + base))[1];
        int4   ga = lg[0];
        int4   gb = lg[1];
        float s[CHUNK] = {sa.x, sa.y, sa.z, sa.w, sb.x, sb.y, sb.z, sb.w};
        int   g[CHUNK] = {ga.x, ga.y, ga.z, ga.w, gb.x, gb.y, gb.z, gb.w};
        float e[CHUNK];

        int i = 0;
        while (i < CHUNK) {
            int cg = g[i];
            float mx = funkey(smax_key[cg]);   // one L2 gather per run
            float part = 0.0f;
            int j = i;
            while (j < CHUNK && g[j] == cg) {
                e[j] = __expf(s[j] - mx);
                part += e[j];
                ++j;
            }
            atomicAdd(&denom64[cg],
                      (unsigned long long)((double)part * FIX_SCALE));
            i = j;
        }
        ((float4*)(e_out + base))[0] = make_float4(e[0], e[1], e[2], e[3]);
        ((float4*)(e_out + base))[1] = make_float4(e[4], e[5], e[6], e[7]);
    } else {
        int end = min(base + CHUNK, E);
        for (int i = base; i < end; ++i) {
            int cg = lds_grp[i - blockBase];
            float e = __expf(s_in[i] - funkey(smax_key[cg]));
            e_out[i] = e;
            atomicAdd(&denom64[cg],
                      (unsigned long long)((double)e * FIX_SCALE));
        }
    }
}

// ---------------------------------------------------------------------------
// Pass 3: soft = e/denom[g]; sn = soft + eps*u_eps; store soft + sn-keys;
//         segment max of sn (keys)
// ---------------------------------------------------------------------------
__global__ void __launch_bounds__(NTHREADS)
k_pass3(const float* __restrict__ e_in,
        const int*   __restrict__ grp,
        const float* __restrict__ ueps,
        const float* __restrict__ denomf,
        float* __restrict__ soft_out,
        unsigned int* __restrict__ snkey_out,
        unsigned int* __restrict__ m_key,
        int E, int use_vec) {
    __shared__ __align__(16) int lds_grp[TILE];
    int blockBase = (int)blockIdx.x * TILE;
    stage_grp(grp, blockBase, E, lds_grp);

    int base = blockBase + (int)threadIdx.x * CHUNK;
    if (base >= E) return;
    const int4* lg = (const int4*)(lds_grp + (int)threadIdx.x * CHUNK);

    if (use_vec && base + CHUNK <= E) {
        float4 ea = ((const float4*)(e_in + base))[0];
        float4 eb = ((const float4*)(e_in + base))[1];
        float4 ua = ((const float4*)(ueps + base))[0];
        float4 ub = ((const float4*)(ueps + base))[1];
        int4   ga = lg[0];
        int4   gb = lg[1];
        float e[CHUNK] = {ea.x, ea.y, ea.z, ea.w, eb.x, eb.y, eb.z, eb.w};
        float u[CHUNK] = {ua.x, ua.y, ua.z, ua.w, ub.x, ub.y, ub.z, ub.w};
        int   g[CHUNK] = {ga.x, ga.y, ga.z, ga.w, gb.x, gb.y, gb.z, gb.w};
        float soft[CHUNK];
        unsigned int key[CHUNK];

        int i = 0;
        while (i < CHUNK) {
            int cg = g[i];
            float rcp = 1.0f / denomf[cg];     // one L2 gather per run
            unsigned int runmax = 0u;
            int j = i;
            while (j < CHUNK && g[j] == cg) {
                soft[j] = e[j] * rcp;
                float sn = soft[j] + EPS_NOISE * u[j];
                key[j] = fkey(sn);
                runmax = max(runmax, key[j]);
                ++j;
            }
            atomicMax(&m_key[cg], runmax);
            i = j;
        }
        ((float4*)(soft_out + base))[0] = make_float4(soft[0], soft[1], soft[2], soft[3]);
        ((float4*)(soft_out + base))[1] = make_float4(soft[4], soft[5], soft[6], soft[7]);
        ((uint4*)(snkey_out + base))[0] = make_uint4(key[0], key[1], key[2], key[3]);
        ((uint4*)(snkey_out + base))[1] = make_uint4(key[4], key[5], key[6], key[7]);
    } else {
        int end = min(base + CHUNK, E);
        for (int i = base; i < end; ++i) {
            int cg = lds_grp[i - blockBase];
            float soft = e_in[i] / denomf[cg];
            float sn = soft + EPS_NOISE * ueps[i];
            unsigned int k = fkey(sn);
            soft_out[i]  = soft;
            snkey_out[i] = k;
            atomicMax(&m_key[cg], k);
        }
    }
}

// ---------------------------------------------------------------------------
// Pass 4: hot = (key == m_key[g]); st = (float)hot.  st overwrites snkey
// in place (no __restrict__ on those two), hot overwrites e region.
// ---------------------------------------------------------------------------
__global__ void __launch_bounds__(NTHREADS)
k_pass4(const unsigned int* snkey,
        const int* __restrict__ grp,
        const unsigned int* __restrict__ m_key,
        float* st_out,                 // aliases snkey
        int* __restrict__ hot_out,
        int E, int use_vec) {
    __shared__ __align__(16) int lds_grp[TILE];
    int blockBase = (int)blockIdx.x * TILE;
    stage_grp(grp, blockBase, E, lds_grp);

    int base = blockBase + (int)threadIdx.x * CHUNK;
    if (base >= E) return;
    const int4* lg = (const int4*)(lds_grp + (int)threadIdx.x * CHUNK);

    if (use_vec && base + CHUNK <= E) {
        uint4 ka = ((const uint4*)(snkey + base))[0];
        uint4 kb = ((const uint4*)(snkey + base))[1];
        int4  ga = lg[0];
        int4  gb = lg[1];
        unsigned int k[CHUNK] = {ka.x, ka.y, ka.z, ka.w, kb.x, kb.y, kb.z, kb.w};
        int g[CHUNK] = {ga.x, ga.y, ga.z, ga.w, gb.x, gb.y, gb.z, gb.w};
        int hot[CHUNK];

        int i = 0;
        while (i < CHUNK) {
            int cg = g[i];
            unsigned int mk = m_key[cg];       // one L2 gather per run
            int j = i;
            while (j < CHUNK && g[j] == cg) { hot[j] = (k[j] == mk) ? 1 : 0; ++j; }
            i = j;
        }
        ((float4*)(st_out + base))[0] =
            make_float4((float)hot[0], (float)hot[1], (float)hot[2], (float)hot[3]);
        ((float4*)(st_out + base))[1] =
            make_float4((float)hot[4], (float)hot[5], (float)hot[6], (float)hot[7]);
        ((int4*)(hot_out + base))[0] = make_int4(hot[0], hot[1], hot[2], hot[3]);
        ((int4*)(hot_out + base))[1] = make_int4(hot[4], hot[5], hot[6], hot[7]);
    } else {
        int end = min(base + CHUNK, E);
        for (int i = base; i < end; ++i) {
            unsigned int k = snkey[i];
            int h = (k == m_key[lds_grp[i - blockBase]]) ? 1 : 0;
            st_out[i]  = (float)h;
            hot_out[i] = h;
        }
    }
}

// ---------------------------------------------------------------------------
// Host launcher
// ---------------------------------------------------------------------------
extern "C" void kernel_launch(void* const* d_in, const int* in_sizes, int n_in,
                              void* d_out, int out_size, void* d_ws, size_t ws_size,
                              hipStream_t stream) {
    const float* logits  = (const float*)d_in[0];
    const int*   grp     = (const int*)d_in[1];
    const int*   ng_dev  = (const int*)d_in[2];   // scalar, device-resident
    const float* u_gum   = (const float*)d_in[3];
    const float* u_eps   = (const float*)d_in[4];
    const int E = in_sizes[0];

    // d_out: [0,E) st | [E,2E) s_hot (int) | [2E,3E) soft.
    // Regions 1/2 double as temporaries: s -> snkey -> st, e -> hot.
    float* reg1 = (float*)d_out;                   // s, then sn-keys, then st
    float* reg2 = (float*)d_out + (size_t)E;       // e, then s_hot
    float* reg3 = (float*)d_out + 2 * (size_t)E;   // soft

    // ws carve: 20 bytes per group (u64 sum + 2 uint keys + f32 denom)
    size_t cap = (ws_size / 20) & ~(size_t)1;
    unsigned long long* denom64  = (unsigned long long*)d_ws;
    unsigned int*       smax_key = (unsigned int*)((char*)d_ws + cap * 8);
    unsigned int*       m_key    = (unsigned int*)((char*)d_ws + cap * 12);
    float*              denomf   = (float*)((char*)d_ws + cap * 16);

    const int use_vec = ((E & 3) == 0) ? 1 : 0;  // region bases 16B-aligned?
    const int chunks  = (E + CHUNK - 1) / CHUNK;
    const int blocks  = (chunks + NTHREADS - 1) / NTHREADS;
    const int gblocks = 2048;                     // grid-stride over n_groups

    k_init<<<gblocks, NTHREADS, 0, stream>>>(smax_key, denom64, m_key, ng_dev);

    k_pass1<<<blocks, NTHREADS, 0, stream>>>(logits, grp, u_gum,
                                             reg1, smax_key, E, use_vec);

    k_pass2<<<blocks, NTHREADS, 0, stream>>>(reg1, grp, smax_key,
                                             reg2, denom64, E, use_vec);

    k_fin_denom<<<gblocks, NTHREADS, 0, stream>>>(denom64, denomf, ng_dev);

    k_pass3<<<blocks, NTHREADS, 0, stream>>>(reg2, grp, u_eps, denomf,
                                             reg3, (unsigned int*)reg1, m_key,
                                             E, use_vec);

    k_pass4<<<blocks, NTHREADS, 0, stream>>>((const unsigned int*)reg1, grp, m_key,
                                             reg1, (int*)reg2, E, use_vec);
}